// PCVRHyFormer_52321291600429
// MI455X (gfx1250) — compile-verified
//
#include <hip/hip_runtime.h>
#include <math.h>
#include <stdint.h>

// ---------------- static config (matches reference) ----------------
#define D_MODEL   512
#define NUM_HEADS 8
#define HEAD_DIM  64
#define L0_LEN    1024
#define L_TOTAL   1536
#define NUM_NS    15
#define T_SEQ     1552            // L_TOTAL + NUM_NS + 1
#define NS0       1536
#define NS1       1551
#define CLS_POS   1551
#define BATCH     4
#define BT        (BATCH * T_SEQ) // 6208  (divisible by 64)
#define HIDDEN    2048
#define NEGINF    (-1e9f)

typedef __attribute__((ext_vector_type(2))) float v2f;
typedef __attribute__((ext_vector_type(8))) float v8f;
typedef int v4i __attribute__((vector_size(16)));

// D = A(16x4 f32) * B(4x16 f32) + C(16x16 f32), full-precision CDNA5 WMMA.
static __device__ __forceinline__ v8f wmma4(v2f a, v2f b, v8f c) {
    return __builtin_amdgcn_wmma_f32_16x16x4_f32(false, a, false, b, (short)0, c,
                                                 false, false);
}

// ---------------- async global->LDS copy (CDNA5 ASYNCcnt path) ----------------
#if defined(__has_builtin)
#if __has_builtin(__builtin_amdgcn_global_load_async_to_lds_b128)
#define USE_ASYNC_LDS 1
#endif
#endif
#ifndef USE_ASYNC_LDS
#define USE_ASYNC_LDS 0
#endif

static __device__ __forceinline__ void cp16(float* dst_lds, const float* src_g) {
#if USE_ASYNC_LDS
    // Generic global addresses are identity-mapped; generic LDS addresses
    // truncate to the 32-bit LDS offset.
    __attribute__((address_space(1))) v4i* gp =
        (__attribute__((address_space(1))) v4i*)(uintptr_t)src_g;
    __attribute__((address_space(3))) v4i* lp =
        (__attribute__((address_space(3))) v4i*)(uint32_t)(uintptr_t)dst_lds;
    __builtin_amdgcn_global_load_async_to_lds_b128(gp, lp, 0, 0);
#else
    *(float4*)dst_lds = *(const float4*)src_g;
#endif
}

static __device__ __forceinline__ void wait_async() {
#if USE_ASYNC_LDS
#if __has_builtin(__builtin_amdgcn_s_wait_asynccnt)
    __builtin_amdgcn_s_wait_asynccnt(0);
#else
    asm volatile("s_wait_asynccnt 0x0" ::: "memory");
#endif
#endif
}

// ---------------- LayerNorm: one wave (32 lanes) per 512-wide row ----------------
__global__ __launch_bounds__(256)
void ln_kernel(const float* __restrict__ x, const float* __restrict__ g,
               const float* __restrict__ b, float* __restrict__ out, int nrows) {
    int wave = threadIdx.x >> 5, lane = threadIdx.x & 31;
    int row = blockIdx.x * 8 + wave;
    if (row >= nrows) return;
    const float* xr = x + (size_t)row * D_MODEL;
    float4 v[4];
    float s = 0.f;
#pragma unroll
    for (int i = 0; i < 4; ++i) {
        v[i] = ((const float4*)xr)[lane * 4 + i];
        s += v[i].x + v[i].y + v[i].z + v[i].w;
    }
#pragma unroll
    for (int m = 16; m >= 1; m >>= 1) s += __shfl_xor(s, m, 32);
    float mu = s * (1.0f / D_MODEL);
    float vs = 0.f;
#pragma unroll
    for (int i = 0; i < 4; ++i) {
        float dx = v[i].x - mu, dy = v[i].y - mu, dz = v[i].z - mu, dw = v[i].w - mu;
        vs += dx * dx + dy * dy + dz * dz + dw * dw;
    }
#pragma unroll
    for (int m = 16; m >= 1; m >>= 1) vs += __shfl_xor(vs, m, 32);
    float rstd = 1.0f / sqrtf(vs * (1.0f / D_MODEL) + 1e-5f);
    float* orow = out + (size_t)row * D_MODEL;
#pragma unroll
    for (int i = 0; i < 4; ++i) {
        int col = lane * 16 + i * 4;
        float4 gg = *(const float4*)&g[col];
        float4 bb = *(const float4*)&b[col];
        float4 o4;
        o4.x = (v[i].x - mu) * rstd * gg.x + bb.x;
        o4.y = (v[i].y - mu) * rstd * gg.y + bb.y;
        o4.z = (v[i].z - mu) * rstd * gg.z + bb.z;
        o4.w = (v[i].w - mu) * rstd * gg.w + bb.w;
        ((float4*)orow)[lane * 4 + i] = o4;
    }
}

// ---------------- Tiled fp32 WMMA GEMM: C = A(MxK) * W(KxN) [+bias] [+resid] ----
// Block tile 64x64, 8 waves (4 rows x 2 col-strips), wave tile 16x32 (2 accs
// sharing one A fragment -> 2 wmma per A-fragment load).
#define GT_M 64
#define GT_N 64
#define GT_K 32

__global__ __launch_bounds__(256)
void gemm_kernel(const float* __restrict__ A, const float* __restrict__ W,
                 const float* __restrict__ bias, const float* __restrict__ resid,
                 float* __restrict__ C, int M, int N, int K) {
    __shared__ float As[GT_M][GT_K + 4];
    __shared__ float Ws[GT_K][GT_N + 4];
    int t = threadIdx.x;
    int wave = t >> 5, lane = t & 31, half = lane >> 4, nn = lane & 15;
    int wy = wave & 3, wx = wave >> 2;               // 4x2 wave grid over 64x64 tile
    int m0 = blockIdx.x * GT_M, n0 = blockIdx.y * GT_N;
    int ar = t >> 3, ac = (t & 7) * 4;               // A coop-load: rows ar, ar+32
    int wr = t >> 4, wc = (t & 15) * 4;              // W coop-load: f4 idx t and t+256
    v8f acc0 = {}, acc1 = {};
    for (int kb = 0; kb < K; kb += GT_K) {
        __syncthreads();
        cp16(&As[ar][ac],      &A[(size_t)(m0 + ar) * K + kb + ac]);
        cp16(&As[ar + 32][ac], &A[(size_t)(m0 + ar + 32) * K + kb + ac]);
        cp16(&Ws[wr][wc],      &W[(size_t)(kb + wr) * N + n0 + wc]);
        cp16(&Ws[wr + 16][wc], &W[(size_t)(kb + wr + 16) * N + n0 + wc]);
        wait_async();
        __syncthreads();
#pragma unroll
        for (int kk = 0; kk < GT_K; kk += 4) {
            v2f a, b0, b1;
            a.x  = As[wy * 16 + nn][kk + 2 * half];
            a.y  = As[wy * 16 + nn][kk + 2 * half + 1];
            b0.x = Ws[kk + 2 * half][wx * 32 + nn];
            b0.y = Ws[kk + 2 * half + 1][wx * 32 + nn];
            b1.x = Ws[kk + 2 * half][wx * 32 + 16 + nn];
            b1.y = Ws[kk + 2 * half + 1][wx * 32 + 16 + nn];
            acc0 = wmma4(a, b0, acc0);
            acc1 = wmma4(a, b1, acc1);
        }
    }
#pragma unroll
    for (int r = 0; r < 8; ++r) {
        int row  = m0 + wy * 16 + r + 8 * half;
        int col0 = n0 + wx * 32 + nn;
        float v0 = acc0[r], v1 = acc1[r];
        if (bias)  { v0 += bias[col0]; v1 += bias[col0 + 16]; }
        if (resid) { v0 += resid[(size_t)row * N + col0];
                     v1 += resid[(size_t)row * N + col0 + 16]; }
        C[(size_t)row * N + col0]      = v0;
        C[(size_t)row * N + col0 + 16] = v1;
    }
}

// ---------------- Fused SwiGLU GEMM: H = silu(A*W1) * (A*W3), 64x64 tile ----------------
__global__ __launch_bounds__(256)
void gemm_swiglu_kernel(const float* __restrict__ A, const float* __restrict__ W1,
                        const float* __restrict__ W3, float* __restrict__ H,
                        int M, int N, int K) {
    __shared__ float As[GT_M][GT_K + 4];
    __shared__ float W1s[GT_K][GT_N + 4];
    __shared__ float W3s[GT_K][GT_N + 4];
    int t = threadIdx.x;
    int wave = t >> 5, lane = t & 31, half = lane >> 4, nn = lane & 15;
    int wy = wave & 3, wx = wave >> 2;
    int m0 = blockIdx.x * GT_M, n0 = blockIdx.y * GT_N;
    int ar = t >> 3, ac = (t & 7) * 4;
    int wr = t >> 4, wc = (t & 15) * 4;
    v8f a10 = {}, a11 = {}, a30 = {}, a31 = {};
    for (int kb = 0; kb < K; kb += GT_K) {
        __syncthreads();
        cp16(&As[ar][ac],       &A[(size_t)(m0 + ar) * K + kb + ac]);
        cp16(&As[ar + 32][ac],  &A[(size_t)(m0 + ar + 32) * K + kb + ac]);
        cp16(&W1s[wr][wc],      &W1[(size_t)(kb + wr) * N + n0 + wc]);
        cp16(&W1s[wr + 16][wc], &W1[(size_t)(kb + wr + 16) * N + n0 + wc]);
        cp16(&W3s[wr][wc],      &W3[(size_t)(kb + wr) * N + n0 + wc]);
        cp16(&W3s[wr + 16][wc], &W3[(size_t)(kb + wr + 16) * N + n0 + wc]);
        wait_async();
        __syncthreads();
#pragma unroll
        for (int kk = 0; kk < GT_K; kk += 4) {
            v2f a, b;
            a.x = As[wy * 16 + nn][kk + 2 * half];
            a.y = As[wy * 16 + nn][kk + 2 * half + 1];
            b.x = W1s[kk + 2 * half][wx * 32 + nn];
            b.y = W1s[kk + 2 * half + 1][wx * 32 + nn];
            a10 = wmma4(a, b, a10);
            b.x = W1s[kk + 2 * half][wx * 32 + 16 + nn];
            b.y = W1s[kk + 2 * half + 1][wx * 32 + 16 + nn];
            a11 = wmma4(a, b, a11);
            b.x = W3s[kk + 2 * half][wx * 32 + nn];
            b.y = W3s[kk + 2 * half + 1][wx * 32 + nn];
            a30 = wmma4(a, b, a30);
            b.x = W3s[kk + 2 * half][wx * 32 + 16 + nn];
            b.y = W3s[kk + 2 * half + 1][wx * 32 + 16 + nn];
            a31 = wmma4(a, b, a31);
        }
    }
#pragma unroll
    for (int r = 0; r < 8; ++r) {
        int row  = m0 + wy * 16 + r + 8 * half;
        int col0 = n0 + wx * 32 + nn;
        float h10 = a10[r], h11 = a11[r];
        float s0 = 1.0f / (1.0f + expf(-h10));
        float s1 = 1.0f / (1.0f + expf(-h11));
        H[(size_t)row * N + col0]      = h10 * s0 * a30[r];
        H[(size_t)row * N + col0 + 16] = h11 * s1 * a31[r];
    }
}

// ---------------- RoPE (in place), one block per token row ----------------
__global__ __launch_bounds__(256)
void rope_kernel(float* __restrict__ buf) {
    int row = blockIdx.x;                 // b*T + t
    int pos = row % T_SEQ;
    int head = threadIdx.x >> 5, i = threadIdx.x & 31;
    float* p = buf + (size_t)row * D_MODEL + head * HEAD_DIM;
    float x1 = p[i], x2 = p[i + 32];
    float ang = (float)pos * expf(-(float)i * (logf(10000.0f) / 32.0f));
    float sv, cv;
    sincosf(ang, &sv, &cv);
    p[i]      = x1 * cv - x2 * sv;        // x*cos + rotate_half(x)*sin
    p[i + 32] = x2 * cv + x1 * sv;
}

// ---------------- OneTrans mask predicate ----------------
static __device__ __forceinline__ bool mask_ok(int q, int k, int sl0, int sl1) {
    bool base;
    if (q < L0_LEN)        base = (k <= q) || (k >= NS0 && k < NS1);
    else if (q < L_TOTAL)  base = (k >= L0_LEN && k <= q) || (k >= NS0 && k < NS1);
    else if (q < NS1)      base = (k < NS1);
    else                   base = (k < NS1) || (k == CLS_POS);
    bool kok = (k < L0_LEN) ? (k < sl0)
             : (k < L_TOTAL) ? ((k - L0_LEN) < sl1) : true;
    return base && kok;
}

// ---------------- Flash attention: wave = 16 q rows, block = 8 q-tiles of one (b,h) ----------------
__global__ __launch_bounds__(256)
void attn_kernel(const float* __restrict__ Q, const float* __restrict__ Kg,
                 const float* __restrict__ V, float* __restrict__ O,
                 const int* __restrict__ sl0p, const int* __restrict__ sl1p) {
    __shared__ float Ks[16][HEAD_DIM + 4];
    __shared__ float Vs[16][HEAD_DIM + 4];
    __shared__ float Ps[8][16][17];
    int t = threadIdx.x, wave = t >> 5, lane = t & 31, half = lane >> 4, nn = lane & 15;
    int b = blockIdx.y >> 3, h = blockIdx.y & 7;
    int qt = blockIdx.x * 8 + wave;
    if (qt > (T_SEQ / 16 - 1)) qt = T_SEQ / 16 - 1;   // clamp: duplicate tile, identical writes
    int q0 = qt * 16;
    int sl0 = sl0p[b], sl1 = sl1p[b];
    size_t bT = (size_t)b * T_SEQ;

    // Q fragments in WMMA A-layout, registers for whole k loop
    const float* qbase = Q + (bT + q0 + nn) * D_MODEL + h * HEAD_DIM + 2 * half;
    v2f qa[16];
#pragma unroll
    for (int ks = 0; ks < 16; ++ks) {
        float2 qq = *(const float2*)(qbase + ks * 4);
        qa[ks].x = qq.x; qa[ks].y = qq.y;
    }

    float mrow[8], lrow[8];
    v8f o[4] = {{}, {}, {}, {}};
#pragma unroll
    for (int r = 0; r < 8; ++r) { mrow[r] = -1e30f; lrow[r] = 0.f; }

    int krow = t >> 4, kc4 = (t & 15) * 4;
    for (int kt = 0; kt < T_SEQ / 16; ++kt) {
        int k0 = kt * 16;
        __syncthreads();
        cp16(&Ks[krow][kc4], &Kg[(bT + k0 + krow) * D_MODEL + h * HEAD_DIM + kc4]);
        cp16(&Vs[krow][kc4], &V[(bT + k0 + krow) * D_MODEL + h * HEAD_DIM + kc4]);
        wait_async();
        __syncthreads();

        // S(16x16) = Q * K^T  (B fragment = K^T, i.e. Ks[n][k])
        v8f s = {};
#pragma unroll
        for (int ks = 0; ks < 16; ++ks) {
            v2f bf;
            bf.x = Ks[nn][ks * 4 + 2 * half];
            bf.y = Ks[nn][ks * 4 + 2 * half + 1];
            s = wmma4(qa[ks], bf, s);
        }

        // mask + online softmax (rows live in 16-lane halves)
        int kcol = k0 + nn;
#pragma unroll
        for (int r = 0; r < 8; ++r) {
            int qrow = q0 + r + 8 * half;
            float sv = mask_ok(qrow, kcol, sl0, sl1) ? s[r] * 0.125f : NEGINF;
            float rmax = sv;
#pragma unroll
            for (int m = 8; m >= 1; m >>= 1) rmax = fmaxf(rmax, __shfl_xor(rmax, m, 32));
            float mnew = fmaxf(mrow[r], rmax);
            float scal = __expf(mrow[r] - mnew);
            float p = __expf(sv - mnew);
            float rsum = p;
#pragma unroll
            for (int m = 8; m >= 1; m >>= 1) rsum += __shfl_xor(rsum, m, 32);
            lrow[r] = lrow[r] * scal + rsum;
            mrow[r] = mnew;
            Ps[wave][r + 8 * half][nn] = p;
#pragma unroll
            for (int nt = 0; nt < 4; ++nt) o[nt][r] *= scal;
        }
        // intra-wave cross-lane LDS RAW on Ps: keep ds ops ordered & un-hoisted
        asm volatile("s_wait_dscnt 0x0" ::: "memory");

        // O(16x64) += P(16x16) * V(16x64); hoist the 4 P A-fragments (reused 4x)
        v2f pa[4];
#pragma unroll
        for (int k2 = 0; k2 < 4; ++k2) {
            pa[k2].x = Ps[wave][nn][k2 * 4 + 2 * half];
            pa[k2].y = Ps[wave][nn][k2 * 4 + 2 * half + 1];
        }
#pragma unroll
        for (int nt = 0; nt < 4; ++nt) {
#pragma unroll
            for (int k2 = 0; k2 < 4; ++k2) {
                v2f bf;
                bf.x = Vs[k2 * 4 + 2 * half][nt * 16 + nn];
                bf.y = Vs[k2 * 4 + 2 * half + 1][nt * 16 + nn];
                o[nt] = wmma4(pa[k2], bf, o[nt]);
            }
        }
    }

#pragma unroll
    for (int nt = 0; nt < 4; ++nt)
#pragma unroll
        for (int r = 0; r < 8; ++r) {
            int qrow = q0 + r + 8 * half;
            O[(bT + qrow) * D_MODEL + h * HEAD_DIM + nt * 16 + nn] = o[nt][r] / lrow[r];
        }
}

// ---------------- elementwise gate multiply ----------------
__global__ __launch_bounds__(256)
void mul_kernel(float* __restrict__ a, const float* __restrict__ g, int n) {
    int i = blockIdx.x * blockDim.x + threadIdx.x;
    if (i < n) a[i] *= g[i];
}

// ---------------- driver ----------------
extern "C" void kernel_launch(void* const* d_in, const int* in_sizes, int n_in,
                              void* d_out, int out_size, void* d_ws, size_t ws_size,
                              hipStream_t stream) {
    (void)in_sizes; (void)n_in; (void)out_size; (void)ws_size;
    const float* x    = (const float*)d_in[0];
    const float* Wq   = (const float*)d_in[1];
    const float* bq   = (const float*)d_in[2];
    const float* Wk   = (const float*)d_in[3];
    const float* bk   = (const float*)d_in[4];
    const float* Wv   = (const float*)d_in[5];
    const float* bv   = (const float*)d_in[6];
    const float* Wo   = (const float*)d_in[7];
    const float* bo   = (const float*)d_in[8];
    const float* Wg   = (const float*)d_in[9];
    const float* bg   = (const float*)d_in[10];
    const float* ln1g = (const float*)d_in[11];
    const float* ln1b = (const float*)d_in[12];
    const float* w1   = (const float*)d_in[13];
    const float* w3   = (const float*)d_in[14];
    const float* w2   = (const float*)d_in[15];
    const float* ln2g = (const float*)d_in[16];
    const float* ln2b = (const float*)d_in[17];
    const int*   sl0  = (const int*)d_in[18];
    const int*   sl1  = (const int*)d_in[19];
    float* out = (float*)d_out;
    float* ws  = (float*)d_ws;

    size_t S1 = (size_t)BT * D_MODEL;
    float* xn = ws;            // LN output (reused for LN2)
    float* Qb = ws + 1 * S1;
    float* Kb = ws + 2 * S1;
    float* Vb = ws + 3 * S1;
    float* Gb = ws + 4 * S1;
    float* Ob = ws + 5 * S1;
    float* Hb = ws + 6 * S1;   // BT x HIDDEN

    dim3 g512(BT / GT_M, D_MODEL / GT_N);
    dim3 gffn(BT / GT_M, HIDDEN / GT_N);

    ln_kernel<<<BT / 8, 256, 0, stream>>>(x, ln1g, ln1b, xn, BT);
    gemm_kernel<<<g512, 256, 0, stream>>>(xn, Wq, bq, nullptr, Qb, BT, D_MODEL, D_MODEL);
    gemm_kernel<<<g512, 256, 0, stream>>>(xn, Wk, bk, nullptr, Kb, BT, D_MODEL, D_MODEL);
    gemm_kernel<<<g512, 256, 0, stream>>>(xn, Wv, bv, nullptr, Vb, BT, D_MODEL, D_MODEL);
    gemm_kernel<<<g512, 256, 0, stream>>>(xn, Wg, bg, nullptr, Gb, BT, D_MODEL, D_MODEL);
    rope_kernel<<<BT, 256, 0, stream>>>(Qb);
    rope_kernel<<<BT, 256, 0, stream>>>(Kb);
    attn_kernel<<<dim3((T_SEQ / 16 + 7) / 8, BATCH * NUM_HEADS), 256, 0, stream>>>(
        Qb, Kb, Vb, Ob, sl0, sl1);
    int nelem = BT * D_MODEL;
    mul_kernel<<<(nelem + 255) / 256, 256, 0, stream>>>(Ob, Gb, nelem);
    // x1 = x + (o*gate)@Wo + bo   -> stored in d_out
    gemm_kernel<<<g512, 256, 0, stream>>>(Ob, Wo, bo, x, out, BT, D_MODEL, D_MODEL);
    ln_kernel<<<BT / 8, 256, 0, stream>>>(out, ln2g, ln2b, xn, BT);
    gemm_swiglu_kernel<<<gffn, 256, 0, stream>>>(xn, w1, w3, Hb, BT, HIDDEN, D_MODEL);
    // out = x1 + H@w2  (residual read/write by same thread)
    gemm_kernel<<<g512, 256, 0, stream>>>(Hb, w2, nullptr, out, out, BT, D_MODEL, HIDDEN);
}